// VQC_Layer_15668040696206
// MI455X (gfx1250) — compile-verified
//
#include <hip/hip_runtime.h>
#include <math.h>

typedef float v2f __attribute__((ext_vector_type(2)));
typedef float v8f __attribute__((ext_vector_type(8)));

#define TPB 256

// ---------------------------------------------------------------------------
// Prep kernel: everything uniform across all 524k rows is computed ONCE.
//   ws[0]      = 0.5 + softplus(angle_scale)
//   ws[8+g*8]  = Rot matrix for gate g = l*4+q : {u00r,u00i,u01r,u01i,
//                                                 u10r,u10i,u11r,u11i}
// Rot(phi,th,om) = [[ep*c, -em*s],[conj(em)*s, conj(ep)*c]],
//   ep = exp(-i(phi+om)/2), em = exp(+i(phi-om)/2)
// ---------------------------------------------------------------------------
__global__ void vqc_prep(const float* __restrict__ angle_scale,
                         const float* __restrict__ theta,
                         float* __restrict__ ws) {
  int t = threadIdx.x;
  if (t == 0) {
    float a = angle_scale[0];
    ws[0] = 0.5f + fmaxf(a, 0.0f) + log1pf(expf(-fabsf(a)));  // stable softplus
  }
  if (t < 8) {
    const float* th3 = theta + t * 3;          // theta[l][q][0..2], g = l*4+q
    float phi = th3[0], th = th3[1], om = th3[2];
    float c = cosf(0.5f * th), s = sinf(0.5f * th);
    float apo = 0.5f * (phi + om), amo = 0.5f * (phi - om);
    float epr = cosf(apo), epi = -sinf(apo);
    float emr = cosf(amo), emi = sinf(amo);
    float* g = ws + 8 + t * 8;
    g[0] =  epr * c;  g[1] =  epi * c;   // u00 = ep*c
    g[2] = -emr * s;  g[3] = -emi * s;   // u01 = -em*s
    g[4] =  emr * s;  g[5] = -emi * s;   // u10 = conj(em)*s
    g[6] =  epr * c;  g[7] = -epi * c;   // u11 = conj(ep)*c
  }
}

// ---------------------------------------------------------------------------
// Device helpers (fully unrolled, constant indices -> state stays in VGPRs)
// Qubit q maps to bit position p = 3-q of the flat amplitude index.
// ---------------------------------------------------------------------------
__device__ __forceinline__ float gelu_exact(float v) {
  return 0.5f * v * (1.0f + erff(v * 0.70710678118654752f));
}

// First Rot gate (qubit 0, p=3) applied to a purely REAL state.
__device__ __forceinline__ void rot_q0_real(const float* __restrict__ pr,
                                            float* __restrict__ re,
                                            float* __restrict__ im,
                                            const float* __restrict__ u) {
#pragma unroll
  for (int i = 0; i < 8; ++i) {
    float a = pr[i], b = pr[i + 8];
    re[i]     = u[0] * a + u[2] * b;
    im[i]     = u[1] * a + u[3] * b;
    re[i + 8] = u[4] * a + u[6] * b;
    im[i + 8] = u[5] * a + u[7] * b;
  }
}

__device__ __forceinline__ void rot_apply(float* __restrict__ re,
                                          float* __restrict__ im,
                                          const float* __restrict__ u,
                                          int p) {
  float u00r = u[0], u00i = u[1], u01r = u[2], u01i = u[3];
  float u10r = u[4], u10i = u[5], u11r = u[6], u11i = u[7];
  int mask = 1 << p;
#pragma unroll
  for (int i = 0; i < 16; ++i) {
    if (i & mask) continue;
    int j = i | mask;
    float ar = re[i], ai = im[i], br = re[j], bi = im[j];
    re[i] = u00r * ar - u00i * ai + u01r * br - u01i * bi;
    im[i] = u00r * ai + u00i * ar + u01r * bi + u01i * br;
    re[j] = u10r * ar - u10i * ai + u11r * br - u11i * bi;
    im[j] = u10r * ai + u10i * ar + u11r * bi + u11i * br;
  }
}

__device__ __forceinline__ void cnot_apply(float* __restrict__ re,
                                           float* __restrict__ im,
                                           int pc, int pt) {
  int cm = 1 << pc, tm = 1 << pt;
#pragma unroll
  for (int i = 0; i < 16; ++i) {
    if ((i & cm) && !(i & tm)) {
      int j = i | tm;
      float tr = re[i]; re[i] = re[j]; re[j] = tr;
      float ti = im[i]; im[i] = im[j]; im[j] = ti;
    }
  }
}

// ---------------------------------------------------------------------------
// Main kernel: 1 thread = 1 row. Each wave (32 rows) does z@W1^T via two
// 16x16 tiles of chained V_WMMA_F32_16X16X4_F32 (K=8 in full f32 precision).
// LDS staging is wave-local -> no barriers needed.
// ---------------------------------------------------------------------------
__global__ void __launch_bounds__(TPB)
vqc_main(const float* __restrict__ x,
         const float* __restrict__ ln1g, const float* __restrict__ ln1b,
         const float* __restrict__ W1,   const float* __restrict__ b1,
         const float* __restrict__ Wq,   const float* __restrict__ bq,
         const float* __restrict__ ws,
         const float* __restrict__ ln2g, const float* __restrict__ ln2b,
         const float* __restrict__ W2,   const float* __restrict__ b2,
         float* __restrict__ out, int N) {
  __shared__ __align__(16) float sz[TPB * 10];  // z rows, stride 10 (pad)
  __shared__ __align__(16) float sg[TPB * 10];  // GEMM result rows

  const int tid  = threadIdx.x;
  const int row  = blockIdx.x * TPB + tid;
  const int ridx = (row < N) ? row : (N - 1);

  // ---- Phase A: load row, LayerNorm1 --------------------------------------
  float z[8];
  {
    const float4 xa = *(const float4*)(x + (size_t)ridx * 8);
    const float4 xb = *(const float4*)(x + (size_t)ridx * 8 + 4);
    z[0] = xa.x; z[1] = xa.y; z[2] = xa.z; z[3] = xa.w;
    z[4] = xb.x; z[5] = xb.y; z[6] = xb.z; z[7] = xb.w;
    float m = 0.f;
#pragma unroll
    for (int d = 0; d < 8; ++d) m += z[d];
    m *= 0.125f;
    float v = 0.f;
#pragma unroll
    for (int d = 0; d < 8; ++d) { float t = z[d] - m; v += t * t; }
    v *= 0.125f;
    float inv = 1.0f / sqrtf(v + 1e-5f);
#pragma unroll
    for (int d = 0; d < 8; ++d) z[d] = (z[d] - m) * inv * ln1g[d] + ln1b[d];
  }
#pragma unroll
  for (int d = 0; d < 8; ++d) sz[tid * 10 + d] = z[d];

  // ---- Phase B: z @ W1^T via f32 WMMA (two 16-row tiles per wave) ---------
  const int lane = tid & 31;
  const int wave = tid >> 5;
  const int h  = lane >> 4;   // K half-select
  const int nl = lane & 15;   // M (for A) / N (for B)

  // A fragments: lane nl holds row (wavebase+nl); VGPR0/1 = K = 2h, 2h+1
  const int r0 = wave * 32 + nl;       // tile 0 rows 0..15 of the wave
  const int r1 = r0 + 16;              // tile 1 rows 16..31
  v2f a00, a01, a10, a11;
  a00.x = sz[r0 * 10 + 2 * h];     a00.y = sz[r0 * 10 + 2 * h + 1];
  a01.x = sz[r0 * 10 + 4 + 2 * h]; a01.y = sz[r0 * 10 + 5 + 2 * h];
  a10.x = sz[r1 * 10 + 2 * h];     a10.y = sz[r1 * 10 + 2 * h + 1];
  a11.x = sz[r1 * 10 + 4 + 2 * h]; a11.y = sz[r1 * 10 + 5 + 2 * h];

  // B fragments: B[k][n] = W1[n][k]; columns n>=8 are zero padding
  v2f bb0; bb0.x = 0.f; bb0.y = 0.f;
  v2f bb1; bb1.x = 0.f; bb1.y = 0.f;
  if (nl < 8) {
    const float* wr = W1 + nl * 8 + 2 * h;
    bb0.x = wr[0]; bb0.y = wr[1];
    bb1.x = wr[4]; bb1.y = wr[5];
  }

  v8f c0 = {};
  c0 = __builtin_amdgcn_wmma_f32_16x16x4_f32(false, a00, false, bb0,
                                             (short)0, c0, false, false);
  c0 = __builtin_amdgcn_wmma_f32_16x16x4_f32(false, a01, false, bb1,
                                             (short)0, c0, false, false);
  v8f c1 = {};
  c1 = __builtin_amdgcn_wmma_f32_16x16x4_f32(false, a10, false, bb0,
                                             (short)0, c1, false, false);
  c1 = __builtin_amdgcn_wmma_f32_16x16x4_f32(false, a11, false, bb1,
                                             (short)0, c1, false, false);

  // D layout: VGPR v -> M = v + 8h, N = nl. Scatter rows back to LDS.
  if (nl < 8) {
#pragma unroll
    for (int v = 0; v < 8; ++v)
      sg[(wave * 32 + v + 8 * h) * 10 + nl] = c0[v];
#pragma unroll
    for (int v = 0; v < 8; ++v)
      sg[(wave * 32 + 16 + v + 8 * h) * 10 + nl] = c1[v];
  }

  // ---- Phase C: bias + exact GELU -----------------------------------------
  float g[8];
#pragma unroll
  for (int d = 0; d < 8; ++d) g[d] = gelu_exact(sg[tid * 10 + d] + b1[d]);

  // ---- Phase D: xq = tanh((g @ Wq^T + bq) * scale) * pi/2 -----------------
  const float scale = ws[0];
  float xq[4];
#pragma unroll
  for (int o = 0; o < 4; ++o) {
    float s = bq[o];
#pragma unroll
    for (int d = 0; d < 8; ++d) s += g[d] * Wq[o * 8 + d];
    xq[o] = tanhf(s * scale) * 1.57079632679489662f;
  }

  // ---- Phase E: 4-qubit VQC statevector sim in registers ------------------
  // RY embedding on |0000> -> real product state.
  float cq[4], sq[4];
#pragma unroll
  for (int q = 0; q < 4; ++q) {
    float sv, cv;
    sincosf(0.5f * xq[q], &sv, &cv);
    cq[q] = cv; sq[q] = sv;
  }
  float p01[4], p23[4], pr[16];
  p01[0] = cq[0] * cq[1]; p01[1] = cq[0] * sq[1];
  p01[2] = sq[0] * cq[1]; p01[3] = sq[0] * sq[1];
  p23[0] = cq[2] * cq[3]; p23[1] = cq[2] * sq[3];
  p23[2] = sq[2] * cq[3]; p23[3] = sq[2] * sq[3];
#pragma unroll
  for (int i = 0; i < 16; ++i) pr[i] = p01[i >> 2] * p23[i & 3];

  const float* RM = ws + 8;  // 8 gates x 8 floats
  float re[16], im[16];
  // Layer 0: Rot on q0 (real-input specialization), then q1..q3
  rot_q0_real(pr, re, im, RM + 0);
  rot_apply(re, im, RM + 8,  2);
  rot_apply(re, im, RM + 16, 1);
  rot_apply(re, im, RM + 24, 0);
  // CNOT ring r=1: (0,1),(1,2),(2,3),(3,0) -> bit pos (3,2),(2,1),(1,0),(0,3)
  cnot_apply(re, im, 3, 2);
  cnot_apply(re, im, 2, 1);
  cnot_apply(re, im, 1, 0);
  cnot_apply(re, im, 0, 3);
  // Layer 1: Rot q0..q3
  rot_apply(re, im, RM + 32, 3);
  rot_apply(re, im, RM + 40, 2);
  rot_apply(re, im, RM + 48, 1);
  rot_apply(re, im, RM + 56, 0);
  // CNOT ring r=2: (0,2),(1,3),(2,0),(3,1) -> (3,1),(2,0),(1,3),(0,2)
  cnot_apply(re, im, 3, 1);
  cnot_apply(re, im, 2, 0);
  cnot_apply(re, im, 1, 3);
  cnot_apply(re, im, 0, 2);

  // <Z_q> from probabilities
  float ev[4] = {0.f, 0.f, 0.f, 0.f};
#pragma unroll
  for (int i = 0; i < 16; ++i) {
    float p = re[i] * re[i] + im[i] * im[i];
    ev[0] += (i & 8) ? -p : p;
    ev[1] += (i & 4) ? -p : p;
    ev[2] += (i & 2) ? -p : p;
    ev[3] += (i & 1) ? -p : p;
  }

  // ---- Phase F: LayerNorm2 + (ev @ W2^T + b2) + exact GELU ----------------
  float m2 = 0.25f * (ev[0] + ev[1] + ev[2] + ev[3]);
  float v2 = 0.f;
#pragma unroll
  for (int q = 0; q < 4; ++q) { float t = ev[q] - m2; v2 += t * t; }
  v2 *= 0.25f;
  float inv2 = 1.0f / sqrtf(v2 + 1e-5f);
  float ln[4];
#pragma unroll
  for (int q = 0; q < 4; ++q)
    ln[q] = (ev[q] - m2) * inv2 * ln2g[q] + ln2b[q];

  float o0, o1, o2, o3;
  float acc[4];
#pragma unroll
  for (int o = 0; o < 4; ++o) {
    float s = b2[o];
#pragma unroll
    for (int q = 0; q < 4; ++q) s += ln[q] * W2[o * 4 + q];
    acc[o] = gelu_exact(s);
  }
  o0 = acc[0]; o1 = acc[1]; o2 = acc[2]; o3 = acc[3];

  if (row < N) {
    *(float4*)(out + (size_t)row * 4) = make_float4(o0, o1, o2, o3);
  }
}

// ---------------------------------------------------------------------------
extern "C" void kernel_launch(void* const* d_in, const int* in_sizes, int n_in,
                              void* d_out, int out_size, void* d_ws, size_t ws_size,
                              hipStream_t stream) {
  const float* x      = (const float*)d_in[0];
  const float* ln1_g  = (const float*)d_in[1];
  const float* ln1_b  = (const float*)d_in[2];
  const float* W1     = (const float*)d_in[3];
  const float* b1     = (const float*)d_in[4];
  const float* Wq     = (const float*)d_in[5];
  const float* bq     = (const float*)d_in[6];
  const float* ascale = (const float*)d_in[7];
  const float* theta  = (const float*)d_in[8];
  const float* ln2_g  = (const float*)d_in[9];
  const float* ln2_b  = (const float*)d_in[10];
  const float* W2     = (const float*)d_in[11];
  const float* b2     = (const float*)d_in[12];
  float* ws  = (float*)d_ws;
  float* out = (float*)d_out;

  const int N = in_sizes[0] / 8;  // B*T rows

  vqc_prep<<<1, 32, 0, stream>>>(ascale, theta, ws);

  const int blocks = (N + TPB - 1) / TPB;
  vqc_main<<<blocks, TPB, 0, stream>>>(x, ln1_g, ln1_b, W1, b1, Wq, bq, ws,
                                       ln2_g, ln2_b, W2, b2, out, N);
}